// SimpleGCN_16312285790333
// MI455X (gfx1250) — compile-verified
//
#include <hip/hip_runtime.h>
#include <hip/hip_bf16.h>

typedef float v2f __attribute__((ext_vector_type(2)));
typedef float v8f __attribute__((ext_vector_type(8)));

#define N_NODES 50000
#define N_EDGES 800000
#define DIM 128
#define APAD 4                    // row pad (floats): stride 132 -> 4-bank skew
#define LDSTRIDE (DIM + APAD)     // 132
#define NSLAB (N_NODES / 16)      // 3125 exact
#define GEMM_BLOCKS 625           // 3125 / 625 = 5 slabs per block, exact
#define GEMM_LDS_BYTES ((128 + 16) * LDSTRIDE * 4)   // Wt[128][132] + As[16][132] = 76032 B

// ---------------------------------------------------------------------------
// Degree / normalization
// ---------------------------------------------------------------------------
__global__ void gcn_init_deg(float* deg, int n) {
    int i = blockIdx.x * blockDim.x + threadIdx.x;
    if (i < n) deg[i] = 1.0f;  // self-loop contributes 1
}

__global__ void gcn_accum_deg(const int* __restrict__ dst, float* deg, int ne) {
    int e = blockIdx.x * blockDim.x + threadIdx.x;
    if (e < ne) atomicAdd(&deg[dst[e]], 1.0f);
}

__global__ void gcn_finalize_dinv(float* deg, int n) {
    int i = blockIdx.x * blockDim.x + threadIdx.x;
    if (i < n) deg[i] = rsqrtf(deg[i]);  // deg >= 1 always (self-loops)
}

// ---------------------------------------------------------------------------
// Dense GEMM via V_WMMA_F32_16X16X4_F32:  C[N,128] = A[N,128] @ W[128,128]
// Persistent blocks (625), 5 row-slabs each. blockDim = 256 (8 waves); wave w
// owns the 16-col tile [w*16,+16). W is staged ONCE per block into LDS,
// transposed (Wt[n][k]) so B fragments are contiguous K-pairs -> ds b64 loads.
// The 16x128 A slab is re-staged per slab. Inner loop is pure LDS -> WMMA.
// Uses dynamic LDS (76 KB) -- CDNA5 WGP allows up to 320 KB per workgroup.
// ---------------------------------------------------------------------------
__global__ void gcn_gemm_wmma_f32(const float* __restrict__ A,
                                  const float* __restrict__ W,
                                  float* __restrict__ C) {
    extern __shared__ float lds[];
    float (*Wt)[LDSTRIDE] = (float(*)[LDSTRIDE])lds;                    // [128][132]
    float (*As)[LDSTRIDE] = (float(*)[LDSTRIDE])(lds + 128 * LDSTRIDE); // [16][132]

    const int tid  = threadIdx.x;
    const int lane = tid & 31;
    const int wave = tid >> 5;
    const int col0 = wave * 16;
    const int m    = lane & 15;      // tile row (A) / tile col (B)
    const int half = lane >> 4;      // K-half selector per ISA fragment layout

    // --- one-time stage of W, transposed: Wt[n][k] = W[k][n] -----------------
    {
        const float4* __restrict__ Wg = (const float4*)W;
        #pragma unroll
        for (int i = tid; i < DIM * (DIM / 4); i += 256) {
            const int r  = i >> 5;         // W row (k), 0..127
            const int c4 = i & 31;         // float4 column block
            const float4 w = Wg[i];
            Wt[c4 * 4 + 0][r] = w.x;
            Wt[c4 * 4 + 1][r] = w.y;
            Wt[c4 * 4 + 2][r] = w.z;
            Wt[c4 * 4 + 3][r] = w.w;
        }
    }

    for (int slab = blockIdx.x; slab < NSLAB; slab += gridDim.x) {
        const int row0 = slab * 16;

        __syncthreads();   // prior compute done (and Wt staged, 1st iter) before As overwrite

        // --- stage 16x128 A slab: 512 float4, 2 per thread, coalesced -------
        {
            const float4* __restrict__ Ag = (const float4*)(A + (size_t)row0 * DIM);
            #pragma unroll
            for (int i = tid; i < 16 * (DIM / 4); i += 256) {
                const int r  = i >> 5;       // row 0..15
                const int c4 = i & 31;       // float4 index within row
                ((float4*)&As[r][0])[c4] = Ag[(size_t)r * (DIM / 4) + c4];
            }
        }
        __syncthreads();

        v8f acc = {};
        #pragma unroll 4
        for (int k = 0; k < DIM; k += 4) {
            const int k0 = k + (half ? 2 : 0);
            const int k1 = k0 + 1;
            v2f a, b;
            a.x = As[m][k0];                 // conflict-free: banks 4m (+2 for half 1)
            a.y = As[m][k1];
            b.x = Wt[col0 + m][k0];          // contiguous K-pair -> ds b64
            b.y = Wt[col0 + m][k1];
            acc = __builtin_amdgcn_wmma_f32_16x16x4_f32(
                /*neg_a=*/false, a, /*neg_b=*/false, b,
                /*c_mod=*/(short)0, acc, /*reuse_a=*/false, /*reuse_b=*/false);
        }

        // D layout: VGPR r -> row (r + half*8), col = lane&15. Unconditional.
        const int rbase = row0 + (half ? 8 : 0);
        float* __restrict__ Cp = C + (size_t)rbase * DIM + col0 + m;
        #pragma unroll
        for (int r = 0; r < 8; ++r) {
            Cp[(size_t)r * DIM] = acc[r];
        }
    }
}

// ---------------------------------------------------------------------------
// Zero a float buffer (float4 granularity; count4 = #float4)
// ---------------------------------------------------------------------------
__global__ void gcn_zero4(float4* buf, size_t count4) {
    size_t i = (size_t)blockIdx.x * blockDim.x + threadIdx.x;
    if (i < count4) buf[i] = make_float4(0.f, 0.f, 0.f, 0.f);
}

// ---------------------------------------------------------------------------
// Edge scatter: out[dst[e]] += dinv[src]*dinv[dst] * xw[src[e]]
// One wave per edge; lane handles 4 contiguous floats (32*4 = 128).
// ---------------------------------------------------------------------------
__global__ void gcn_scatter_edges(const float* __restrict__ xw,
                                  const int* __restrict__ src,
                                  const int* __restrict__ dst,
                                  const float* __restrict__ dinv,
                                  float* __restrict__ out,
                                  int nedges) {
    const int wv   = (int)(((size_t)blockIdx.x * blockDim.x + threadIdx.x) >> 5);
    const int lane = threadIdx.x & 31;
    if (wv >= nedges) return;
    const int s = src[wv];
    const int d = dst[wv];
    const float nrm = dinv[s] * dinv[d];
    const float4 v = ((const float4*)(xw + (size_t)s * DIM))[lane];
    float* drow = out + (size_t)d * DIM + lane * 4;
    atomicAdd(drow + 0, nrm * v.x);
    atomicAdd(drow + 1, nrm * v.y);
    atomicAdd(drow + 2, nrm * v.z);
    atomicAdd(drow + 3, nrm * v.w);
}

// ---------------------------------------------------------------------------
// Self-loop contribution + bias + ReLU (elementwise over [N,128])
// out[i,c] = relu(out[i,c] + dinv[i]^2 * xw[i,c] + b[c])
// ---------------------------------------------------------------------------
__global__ void gcn_selfloop_bias_relu(const float* __restrict__ xw,
                                       const float* __restrict__ dinv,
                                       const float* __restrict__ b,
                                       float* __restrict__ out,
                                       int nnodes) {
    size_t idx = (size_t)blockIdx.x * blockDim.x + threadIdx.x;
    const size_t total = (size_t)nnodes * DIM;
    if (idx >= total) return;
    const int node = (int)(idx >> 7);
    const int col  = (int)(idx & (DIM - 1));
    const float di = dinv[node];
    const float v  = out[idx] + di * di * xw[idx] + b[col];
    out[idx] = fmaxf(v, 0.0f);
}

// ---------------------------------------------------------------------------
// Head: out[n, 0:2] = h[n, :] @ Wl[128,2] + bl.  One wave per node, lane-strided
// partial dots reduced with wave32 shuffles.
// ---------------------------------------------------------------------------
__global__ void gcn_head(const float* __restrict__ h,
                         const float* __restrict__ Wl,
                         const float* __restrict__ bl,
                         float* __restrict__ out,
                         int nnodes) {
    const int wv   = (int)(((size_t)blockIdx.x * blockDim.x + threadIdx.x) >> 5);
    const int lane = threadIdx.x & 31;
    if (wv >= nnodes) return;
    const float* row = h + (size_t)wv * DIM;
    float a0 = 0.f, a1 = 0.f;
    #pragma unroll
    for (int k = lane; k < DIM; k += 32) {
        const float x = row[k];
        a0 += x * Wl[k * 2 + 0];
        a1 += x * Wl[k * 2 + 1];
    }
    #pragma unroll
    for (int off = 16; off > 0; off >>= 1) {
        a0 += __shfl_down(a0, off, 32);
        a1 += __shfl_down(a1, off, 32);
    }
    if (lane == 0) {
        out[(size_t)wv * 2 + 0] = a0 + bl[0];
        out[(size_t)wv * 2 + 1] = a1 + bl[1];
    }
}

// ---------------------------------------------------------------------------
// Launcher
// ---------------------------------------------------------------------------
extern "C" void kernel_launch(void* const* d_in, const int* in_sizes, int n_in,
                              void* d_out, int out_size, void* d_ws, size_t ws_size,
                              hipStream_t stream) {
    const float* x   = (const float*)d_in[0];
    const int*   ei  = (const int*)d_in[1];     // [2, E] flat
    const float* W1  = (const float*)d_in[2];
    const float* b1  = (const float*)d_in[3];
    const float* W2  = (const float*)d_in[4];
    const float* b2  = (const float*)d_in[5];
    const float* Wl  = (const float*)d_in[6];
    const float* bl  = (const float*)d_in[7];
    float* out = (float*)d_out;

    const int* src = ei;
    const int* dst = ei + N_EDGES;

    // Workspace layout (256B-aligned slabs)
    float* ws   = (float*)d_ws;
    const size_t dinv_pad = 50048;                 // 50000 rounded up, 256B aligned
    float* dinv = ws;                              // [N]
    float* bufA = ws + dinv_pad;                   // [N,128] GEMM output
    float* bufB = bufA + (size_t)N_NODES * DIM;    // [N,128] aggregation / activations

    const int T = 256;
    const int feat4   = (int)(((size_t)N_NODES * DIM / 4 + T - 1) / T);  // zero grid
    const int featAll = (int)(((size_t)N_NODES * DIM + T - 1) / T);      // elementwise grid
    const int nodeG   = (N_NODES + T - 1) / T;
    const int edgeG   = (N_EDGES + T - 1) / T;
    const int edgeWvG = (int)(((size_t)N_EDGES * 32 + T - 1) / T);       // wave/edge
    const int nodeWvG = (int)(((size_t)N_NODES * 32 + T - 1) / T);       // wave/node

    // ---- normalization: deg -> dinv ----
    gcn_init_deg<<<nodeG, T, 0, stream>>>(dinv, N_NODES);
    gcn_accum_deg<<<edgeG, T, 0, stream>>>(dst, dinv, N_EDGES);
    gcn_finalize_dinv<<<nodeG, T, 0, stream>>>(dinv, N_NODES);

    // ---- layer 1 ----
    gcn_gemm_wmma_f32<<<GEMM_BLOCKS, T, GEMM_LDS_BYTES, stream>>>(x, W1, bufA);
    gcn_zero4<<<feat4, T, 0, stream>>>((float4*)bufB, (size_t)N_NODES * DIM / 4);
    gcn_scatter_edges<<<edgeWvG, T, 0, stream>>>(bufA, src, dst, dinv, bufB, N_EDGES);
    gcn_selfloop_bias_relu<<<featAll, T, 0, stream>>>(bufA, dinv, b1, bufB, N_NODES);

    // ---- layer 2 ----
    gcn_gemm_wmma_f32<<<GEMM_BLOCKS, T, GEMM_LDS_BYTES, stream>>>(bufB, W2, bufA);
    gcn_zero4<<<feat4, T, 0, stream>>>((float4*)bufB, (size_t)N_NODES * DIM / 4);
    gcn_scatter_edges<<<edgeWvG, T, 0, stream>>>(bufA, src, dst, dinv, bufB, N_EDGES);
    gcn_selfloop_bias_relu<<<featAll, T, 0, stream>>>(bufA, dinv, b2, bufB, N_NODES);

    // ---- head ----
    gcn_head<<<nodeWvG, T, 0, stream>>>(bufB, Wl, bl, out, N_NODES);
}